// MultiRelGraphLayer_42898133352616
// MI455X (gfx1250) — compile-verified
//
#include <hip/hip_runtime.h>

typedef float v8f __attribute__((ext_vector_type(8)));
typedef float v4f __attribute__((ext_vector_type(4)));
typedef float v2f __attribute__((ext_vector_type(2)));

constexpr int   D        = 128;     // node/edge feature dim
constexpr int   H        = 128;     // output dim
constexpr int   KE       = 256;     // 2*D (concat input dim of W_neigh)
constexpr int   N_NODES  = 100000;
constexpr int   N_EDGES  = 600000;
constexpr float RRELU_SLOPE = 0.22916666666666666f; // (1/8 + 1/3)/2

// LDS strides padded so (stride % 64) == 4 -> conflict-free WMMA frag reads
constexpr int KSTR_E = KE + 4;  // 260
constexpr int KSTR_N = D + 4;   // 132

constexpr int EDGE_W_FLOATS    = H * KSTR_E;                            // 33280
constexpr int EDGE_A_FLOATS    = 16 * KSTR_E;                           // per-wave A tile
constexpr int EDGE_SMEM_FLOATS = EDGE_W_FLOATS + H + 8 * EDGE_A_FLOATS; // ~260.5 KB

constexpr int NODE_W_FLOATS    = H * KSTR_N;                            // 16896
constexpr int NODE_A_FLOATS    = 16 * KSTR_N;
constexpr int NODE_SMEM_FLOATS = NODE_W_FLOATS + H + 8 * NODE_A_FLOATS; // ~132.5 KB

// ---------------------------------------------------------------------------
// Zero the accumulation workspace (agg + cnt) every launch.
// ---------------------------------------------------------------------------
__global__ void zero_ws_kernel(v4f* __restrict__ p, int n4) {
    const v4f z = {0.f, 0.f, 0.f, 0.f};
    for (int i = blockIdx.x * blockDim.x + threadIdx.x; i < n4;
         i += gridDim.x * blockDim.x)
        p[i] = z;
}

// ---------------------------------------------------------------------------
// Prefetch helpers: cover one 16-row tile with global_prefetch_b8.
// Each prefetch instruction covers 32 lanes x 1 cacheline (128B granularity).
// ---------------------------------------------------------------------------
__device__ __forceinline__ void prefetch_edge_tile(
    const float* __restrict__ node_feats, const float* __restrict__ edge_feats,
    const int* __restrict__ src, int ebase, int lane) {
    const int q = lane >> 2;     // 0..7 -> row within half-tile
    const int c = lane & 3;      // 0..3 -> 128B chunk within 512B row
    const int s0 = src[ebase + q];
    const int s1 = src[ebase + 8 + q];
    __builtin_prefetch(node_feats + s0 * D + c * 32, 0, 3);
    __builtin_prefetch(node_feats + s1 * D + c * 32, 0, 3);
    // edge rows of one tile are a contiguous 8KB block
    const float* eb = edge_feats + ebase * D;
    __builtin_prefetch(eb + lane * 32, 0, 1);
    __builtin_prefetch(eb + 1024 + lane * 32, 0, 1);
}

__device__ __forceinline__ void prefetch_node_tile(
    const float* __restrict__ node_feats, int nbase, int lane) {
    const float* nb = node_feats + nbase * D;   // contiguous 8KB
    __builtin_prefetch(nb + lane * 32, 0, 1);
    __builtin_prefetch(nb + 1024 + lane * 32, 0, 1);
}

// ---------------------------------------------------------------------------
// Edge kernel: per 16-edge tile, A = [node_feats[src] || edge_feats] (16x256),
// B = W_neigh (256x128), C += A@B via v_wmma_f32_16x16x4_f32, then bias-add
// and atomic scatter-add into agg[dst], plus cnt[dst] += 1.
// Next tile is covered by global_prefetch_b8 issued before the WMMA phase.
// ---------------------------------------------------------------------------
__global__ __launch_bounds__(256) void edge_msg_kernel(
    const float* __restrict__ node_feats,   // [N_NODES, 128]
    const float* __restrict__ edge_feats,   // [N_EDGES, 128] (streamed, NT)
    const float* __restrict__ Wn,           // [256, 128] row-major
    const float* __restrict__ bn,           // [128]
    const int*   __restrict__ src,
    const int*   __restrict__ dst,
    float*       __restrict__ agg,          // [N_NODES, 128]
    float*       __restrict__ cnt)          // [N_NODES]
{
    extern __shared__ float smem[];
    float* sW    = smem;                    // transposed W: [n][k], stride KSTR_E
    float* sBias = smem + EDGE_W_FLOATS;

    const int tid = threadIdx.x;

    // Stage W_neigh transposed: sW[n*KSTR_E + k] = Wn[k*128 + n]
    for (int i = tid; i < KE * H; i += 256) {
        const int k = i >> 7;       // row of W (K index)
        const int n = i & 127;      // col of W (N index)
        sW[n * KSTR_E + k] = Wn[i];
    }
    if (tid < H) sBias[tid] = bn[tid];
    __syncthreads();

    const int lane = tid & 31;
    const int wave = tid >> 5;
    const int l15  = lane & 15;
    const int hh   = lane >> 4;     // half-select (K offset 0 or 2; M offset 0 or 8)
    float* sA = smem + EDGE_W_FLOATS + H + wave * EDGE_A_FLOATS;

    float biasv[8];
#pragma unroll
    for (int t = 0; t < 8; ++t) biasv[t] = sBias[t * 16 + l15];

    const int n_tiles = N_EDGES / 16;            // 37500
    const int gwave   = blockIdx.x * 8 + wave;
    const int nwaves  = gridDim.x * 8;

    if (gwave < n_tiles)
        prefetch_edge_tile(node_feats, edge_feats, src, gwave * 16, lane);

    for (int tile = gwave; tile < n_tiles; tile += nwaves) {
        const int ebase = tile * 16;

        // Commit A tile to LDS: row r = [node_feats[src[e]] | edge_feats[e]]
        // (loads should hit WGP$/L2 thanks to the prefetch issued last iter)
#pragma unroll
        for (int r = 0; r < 16; ++r) {
            const int e = ebase + r;
            const v4f vn = *(const v4f*)(node_feats + src[e] * D + lane * 4);
            const v4f ve = __builtin_nontemporal_load(
                               (const v4f*)(edge_feats + e * D + lane * 4));
            *(v4f*)(sA + r * KSTR_E + lane * 4)     = vn;
            *(v4f*)(sA + r * KSTR_E + D + lane * 4) = ve;
        }
        asm volatile("s_wait_dscnt 0" ::: "memory");

        // Prefetch the NEXT tile (no registers held, no counters touched)
        const int nxt = tile + nwaves;
        if (nxt < n_tiles)
            prefetch_edge_tile(node_feats, edge_feats, src, nxt * 16, lane);

        v8f acc[8] = {};
        const float* aRow  = sA + l15 * KSTR_E + 2 * hh;
        const float* bBase = sW + l15 * KSTR_E + 2 * hh;
#pragma unroll 4
        for (int kk = 0; kk < KE / 4; ++kk) {
            const int kb = kk * 4;
            const v2f a = *(const v2f*)(aRow + kb);
#pragma unroll
            for (int t = 0; t < 8; ++t) {
                const v2f b = *(const v2f*)(bBase + t * 16 * KSTR_E + kb);
                acc[t] = __builtin_amdgcn_wmma_f32_16x16x4_f32(
                    false, a, false, b, (short)0, acc[t], false, false);
            }
        }

        // Scatter: accum[j] holds edge row M = j + 8*hh, column N = t*16 + l15
#pragma unroll
        for (int j = 0; j < 8; ++j) {
            const int e     = ebase + j + 8 * hh;
            const int dnode = dst[e];
            float* row = agg + dnode * H;
#pragma unroll
            for (int t = 0; t < 8; ++t)
                atomicAdd(row + t * 16 + l15, acc[t][j] + biasv[t]);
        }
        if (lane < 16) atomicAdd(cnt + dst[ebase + lane], 1.0f);
    }
}

// ---------------------------------------------------------------------------
// Node kernel: out = rrelu(agg/max(cnt,1) + node_feats @ W_loop + b_loop)
// ---------------------------------------------------------------------------
__global__ __launch_bounds__(256) void node_out_kernel(
    const float* __restrict__ node_feats,   // [N_NODES, 128]
    const float* __restrict__ Wl,           // [128, 128]
    const float* __restrict__ bl,           // [128]
    const float* __restrict__ agg,
    const float* __restrict__ cnt,
    float*       __restrict__ out)          // [N_NODES, 128]
{
    extern __shared__ float smem[];
    float* sW    = smem;                    // transposed W_loop: [n][k], stride KSTR_N
    float* sBias = smem + NODE_W_FLOATS;

    const int tid = threadIdx.x;
    for (int i = tid; i < D * H; i += 256) {
        const int k = i >> 7;
        const int n = i & 127;
        sW[n * KSTR_N + k] = Wl[i];
    }
    if (tid < H) sBias[tid] = bl[tid];
    __syncthreads();

    const int lane = tid & 31;
    const int wave = tid >> 5;
    const int l15  = lane & 15;
    const int hh   = lane >> 4;
    float* sA = smem + NODE_W_FLOATS + H + wave * NODE_A_FLOATS;

    float biasv[8];
#pragma unroll
    for (int t = 0; t < 8; ++t) biasv[t] = sBias[t * 16 + l15];

    const int n_tiles = N_NODES / 16;            // 6250
    const int gwave   = blockIdx.x * 8 + wave;
    const int nwaves  = gridDim.x * 8;

    if (gwave < n_tiles)
        prefetch_node_tile(node_feats, gwave * 16, lane);

    for (int tile = gwave; tile < n_tiles; tile += nwaves) {
        const int nbase = tile * 16;

#pragma unroll
        for (int r = 0; r < 16; ++r) {
            const v4f v = *(const v4f*)(node_feats + (nbase + r) * D + lane * 4);
            *(v4f*)(sA + r * KSTR_N + lane * 4) = v;
        }
        asm volatile("s_wait_dscnt 0" ::: "memory");

        const int nxt = tile + nwaves;
        if (nxt < n_tiles)
            prefetch_node_tile(node_feats, nxt * 16, lane);

        v8f acc[8] = {};
        const float* aRow  = sA + l15 * KSTR_N + 2 * hh;
        const float* bBase = sW + l15 * KSTR_N + 2 * hh;
#pragma unroll 4
        for (int kk = 0; kk < D / 4; ++kk) {
            const int kb = kk * 4;
            const v2f a = *(const v2f*)(aRow + kb);
#pragma unroll
            for (int t = 0; t < 8; ++t) {
                const v2f b = *(const v2f*)(bBase + t * 16 * KSTR_N + kb);
                acc[t] = __builtin_amdgcn_wmma_f32_16x16x4_f32(
                    false, a, false, b, (short)0, acc[t], false, false);
            }
        }

        // Fused epilogue: mean-aggregate + bias + RReLU(eval)
#pragma unroll
        for (int j = 0; j < 8; ++j) {
            const int node = nbase + j + 8 * hh;
            const float c   = cnt[node];
            const float inv = 1.0f / fmaxf(c, 1.0f);
            const float* arow = agg + node * H;
            float*       orow = out + node * H;
#pragma unroll
            for (int t = 0; t < 8; ++t) {
                const int col = t * 16 + l15;
                const float aval = __builtin_nontemporal_load(arow + col);
                float v = acc[t][j] + biasv[t] + aval * inv;
                v = (v >= 0.0f) ? v : RRELU_SLOPE * v;
                __builtin_nontemporal_store(v, orow + col);
            }
        }
    }
}

// ---------------------------------------------------------------------------
extern "C" void kernel_launch(void* const* d_in, const int* in_sizes, int n_in,
                              void* d_out, int out_size, void* d_ws, size_t ws_size,
                              hipStream_t stream) {
    (void)in_sizes; (void)n_in; (void)out_size; (void)ws_size;
    const float* node_feats = (const float*)d_in[0];
    const float* edge_feats = (const float*)d_in[1];
    const float* Wn         = (const float*)d_in[2];
    const float* bn         = (const float*)d_in[3];
    const float* Wl         = (const float*)d_in[4];
    const float* bl         = (const float*)d_in[5];
    const int*   src        = (const int*)d_in[6];
    const int*   dst        = (const int*)d_in[7];
    float* out = (float*)d_out;

    float* agg = (float*)d_ws;                       // [N_NODES * 128]
    float* cnt = agg + (size_t)N_NODES * H;          // [N_NODES]

    // Opt in to >64KB dynamic LDS (host-side, capture-safe, deterministic)
    (void)hipFuncSetAttribute((const void*)edge_msg_kernel,
                              hipFuncAttributeMaxDynamicSharedMemorySize,
                              EDGE_SMEM_FLOATS * (int)sizeof(float));
    (void)hipFuncSetAttribute((const void*)node_out_kernel,
                              hipFuncAttributeMaxDynamicSharedMemorySize,
                              NODE_SMEM_FLOATS * (int)sizeof(float));

    // 1) zero agg + cnt  (12,900,000 floats, divisible by 4)
    const int n4 = (N_NODES * H + N_NODES) / 4;
    zero_ws_kernel<<<2048, 256, 0, stream>>>((v4f*)d_ws, n4);

    // 2) edge messages + mean-aggregate numerator/denominator
    edge_msg_kernel<<<1024, 256, EDGE_SMEM_FLOATS * sizeof(float), stream>>>(
        node_feats, edge_feats, Wn, bn, src, dst, agg, cnt);

    // 3) self-loop GEMM + fused mean + bias + RReLU
    node_out_kernel<<<512, 256, NODE_SMEM_FLOATS * sizeof(float), stream>>>(
        node_feats, Wl, bl, agg, cnt, out);
}